// MutagGCN_Explainer_87041807221076
// MI455X (gfx1250) — compile-verified
//
#include <hip/hip_runtime.h>
#include <hip/hip_bf16.h>

typedef float v2f __attribute__((ext_vector_type(2)));
typedef float v8f __attribute__((ext_vector_type(8)));

// ---------------------------------------------------------------------------
// Workspace layout (floats):
//   [0,32)                  gmax (unsigned bits, init 0 == 0.0f)
//   [32, 32+N)              deg
//   [32+N, 32+13N)          acc1  (N x 12, column 11 stays zero)
//   [32+13N, 32+45N)        acc2  (N x 32)
//   [32+45N, 32+77N)        h1    (N x 32)
// Zero region: first 32 + 45N floats.
// ---------------------------------------------------------------------------

__device__ __forceinline__ void atomAddF(float* p, float v) {
    unsafeAtomicAdd(p, v);   // native global_atomic_add_f32 on gfx1250
}

__device__ __forceinline__ float sigmoidf(float x) {
    return 1.0f / (1.0f + __expf(-x));
}

// ------------------------- zero init ---------------------------------------
__global__ void zero_kernel(float* __restrict__ p, long long n) {
    long long i = (long long)blockIdx.x * blockDim.x + threadIdx.x;
    long long stride = (long long)gridDim.x * blockDim.x;
    for (; i < n; i += stride) p[i] = 0.0f;
}

// ------------------------- layer-1 edge scatter ----------------------------
// Warp stages 32 edges with coalesced index/mask loads, then processes two
// edges per step: lanes 0-15 handle edge i, lanes 16-31 handle edge i+1.
// Slot j<4 -> edge_feat, 4<=j<11 -> node_feat[src], j==11 -> deg count.
__global__ void edge1_kernel(const float* __restrict__ node_feat,
                             const float* __restrict__ edge_feat,
                             const float* __restrict__ edge_mask,
                             const int*   __restrict__ src,
                             const int*   __restrict__ dst,
                             float* __restrict__ acc1,
                             float* __restrict__ deg,
                             int nEdges) {
    const int warpId = (blockIdx.x * blockDim.x + threadIdx.x) >> 5;
    const int lane   = threadIdx.x & 31;
    const int nWarps = (gridDim.x * blockDim.x) >> 5;
    const int halfi  = lane >> 4;     // which edge of the pair
    const int j      = lane & 15;     // feature slot

    for (int base = warpId * 32; base < nEdges; base += nWarps * 32) {
        int e_l = base + lane;
        int e_c = e_l < nEdges ? e_l : (nEdges - 1);
        int   s_l = src[e_c];
        int   d_l = dst[e_c];
        float m_l = sigmoidf(edge_mask[e_c >> 1]);

        for (int i = 0; i < 32; i += 2) {
            int idx = i + halfi;
            int e   = base + idx;
            if (e < nEdges) {
                int   s = __shfl(s_l, idx, 32);
                int   d = __shfl(d_l, idx, 32);
                float m = __shfl(m_l, idx, 32);
                if (j < 11) {
                    float val = (j < 4) ? edge_feat[(long long)e * 4 + j]
                                        : node_feat[(long long)s * 7 + (j - 4)];
                    atomAddF(&acc1[(long long)d * 12 + j], m * val);
                } else if (j == 11) {
                    atomAddF(&deg[d], 1.0f);
                }
            }
        }
    }
}

// ------------------------- layer-2 edge scatter ----------------------------
// Warp stages 32 edges, then one edge per step with all 32 lanes active:
// lane = feature. Gathers and atomics are coalesced 128B transactions; acc2
// (25.6 MB) and h1 (25.6 MB) are L2-resident on the 192 MB L2.
__global__ void edge2_kernel(const float* __restrict__ h1,
                             const float* __restrict__ edge_mask,
                             const int*   __restrict__ src,
                             const int*   __restrict__ dst,
                             float* __restrict__ acc2,
                             int nEdges) {
    const int warpId = (blockIdx.x * blockDim.x + threadIdx.x) >> 5;
    const int lane   = threadIdx.x & 31;
    const int nWarps = (gridDim.x * blockDim.x) >> 5;

    for (int base = warpId * 32; base < nEdges; base += nWarps * 32) {
        int e_l = base + lane;
        int e_c = e_l < nEdges ? e_l : (nEdges - 1);
        int   s_l = src[e_c];
        int   d_l = dst[e_c];
        float m_l = sigmoidf(edge_mask[e_c >> 1]);

        int iMax = nEdges - base;
        if (iMax > 32) iMax = 32;
        for (int i = 0; i < iMax; ++i) {
            int   s = __shfl(s_l, i, 32);
            int   d = __shfl(d_l, i, 32);
            float m = __shfl(m_l, i, 32);
            float val = m * h1[(long long)s * 32 + lane];
            atomAddF(&acc2[(long long)d * 32 + lane], val);
        }
    }
}

// ------------------------- MLP1 (WMMA f32) ---------------------------------
// One wave owns a 16-node tile: h1 = relu(relu((acc1/deg)@W1a+b1a)@W1b+b1b)
// acc1 has stride 12 with a zero column 11; W1a is staged into LDS padded to
// 12x32 with a zero row -> GEMM1 loads are unconditional (no EXEC churn).
__global__ void mlp1_kernel(const float* __restrict__ acc1,
                            const float* __restrict__ deg,
                            const float* __restrict__ W1a, const float* __restrict__ b1a,
                            const float* __restrict__ W1b, const float* __restrict__ b1b,
                            float* __restrict__ h1, int nNodes) {
    __shared__ float W1as[12][32];
    __shared__ float Y[8][16][33];
    const int lane = threadIdx.x & 31;
    const int wave = threadIdx.x >> 5;
    const int half = lane >> 4;      // 0|1 : which half-wave
    const int r    = lane & 15;      // row within tile / column index
    const int base = (blockIdx.x * 8 + wave) * 16;

    {   // stage padded W1a
        int t = threadIdx.x;
        if (t < 12 * 32) {
            int row = t >> 5, col = t & 31;
            W1as[row][col] = (row < 11) ? W1a[row * 32 + col] : 0.0f;
        }
    }
    __syncthreads();

    const int  rowg  = base + r;
    const bool valid = rowg < nNodes;
    const int  rowc  = valid ? rowg : 0;
    const float inv  = valid ? (1.0f / fmaxf(deg[rowc], 1.0f)) : 0.0f;

    // GEMM1: [16x12] @ [12x32], 3 k-steps of 4
    v8f c0 = {}; v8f c1 = {};
#pragma unroll
    for (int t = 0; t < 3; ++t) {
        int k0 = 4 * t + 2 * half;
        v2f a, b0, b1;
        a.x  = acc1[(long long)rowc * 12 + k0]     * inv;
        a.y  = acc1[(long long)rowc * 12 + k0 + 1] * inv;
        b0.x = W1as[k0][r];          b0.y = W1as[k0 + 1][r];
        b1.x = W1as[k0][16 + r];     b1.y = W1as[k0 + 1][16 + r];
        c0 = __builtin_amdgcn_wmma_f32_16x16x4_f32(false, a, false, b0, (short)0, c0, false, false);
        c1 = __builtin_amdgcn_wmma_f32_16x16x4_f32(false, a, false, b1, (short)0, c1, false, false);
    }
    // bias + relu, park in LDS to re-stripe C/D layout -> A-fragment layout
    float bias0 = b1a[r], bias1 = b1a[16 + r];
#pragma unroll
    for (int v = 0; v < 8; ++v) {
        int M = v + 8 * half;
        Y[wave][M][r]      = fmaxf(c0[v] + bias0, 0.0f);
        Y[wave][M][16 + r] = fmaxf(c1[v] + bias1, 0.0f);
    }
    __syncthreads();

    // GEMM2: [16x32] @ [32x32], 8 k-steps
    v8f d0 = {}; v8f d1 = {};
#pragma unroll
    for (int t = 0; t < 8; ++t) {
        int k0 = 4 * t + 2 * half;
        v2f a, b0, b1;
        a.x  = Y[wave][r][k0];
        a.y  = Y[wave][r][k0 + 1];
        b0.x = W1b[k0 * 32 + r];       b0.y = W1b[(k0 + 1) * 32 + r];
        b1.x = W1b[k0 * 32 + 16 + r];  b1.y = W1b[(k0 + 1) * 32 + 16 + r];
        d0 = __builtin_amdgcn_wmma_f32_16x16x4_f32(false, a, false, b0, (short)0, d0, false, false);
        d1 = __builtin_amdgcn_wmma_f32_16x16x4_f32(false, a, false, b1, (short)0, d1, false, false);
    }
    float bias0b = b1b[r], bias1b = b1b[16 + r];
#pragma unroll
    for (int v = 0; v < 8; ++v) {
        int M = v + 8 * half;
        int node = base + M;
        if (node < nNodes) {
            h1[(long long)node * 32 + r]      = fmaxf(d0[v] + bias0b, 0.0f);
            h1[(long long)node * 32 + 16 + r] = fmaxf(d1[v] + bias1b, 0.0f);
        }
    }
}

// ------------------------- MLP2 + global max-pool (WMMA f32) ---------------
__global__ void mlp2_kernel(const float* __restrict__ acc2,
                            const float* __restrict__ deg,
                            const float* __restrict__ W2a, const float* __restrict__ b2a,
                            const float* __restrict__ W2b, const float* __restrict__ b2b,
                            unsigned* __restrict__ gmax, int nNodes) {
    __shared__ float Y[8][16][33];
    __shared__ unsigned smax[32];
    const int lane = threadIdx.x & 31;
    const int wave = threadIdx.x >> 5;
    const int half = lane >> 4;
    const int r    = lane & 15;
    const int base = (blockIdx.x * 8 + wave) * 16;

    if (threadIdx.x < 32) smax[threadIdx.x] = 0u;

    const int  rowg  = base + r;
    const bool valid = rowg < nNodes;
    const int  rowc  = valid ? rowg : 0;
    const float inv  = valid ? (1.0f / fmaxf(deg[rowc], 1.0f)) : 0.0f;

    // GEMM1: [16x32] @ [32x32]
    v8f c0 = {}; v8f c1 = {};
#pragma unroll
    for (int t = 0; t < 8; ++t) {
        int k0 = 4 * t + 2 * half;
        v2f a, b0, b1;
        a.x  = acc2[(long long)rowc * 32 + k0]     * inv;
        a.y  = acc2[(long long)rowc * 32 + k0 + 1] * inv;
        b0.x = W2a[k0 * 32 + r];       b0.y = W2a[(k0 + 1) * 32 + r];
        b1.x = W2a[k0 * 32 + 16 + r];  b1.y = W2a[(k0 + 1) * 32 + 16 + r];
        c0 = __builtin_amdgcn_wmma_f32_16x16x4_f32(false, a, false, b0, (short)0, c0, false, false);
        c1 = __builtin_amdgcn_wmma_f32_16x16x4_f32(false, a, false, b1, (short)0, c1, false, false);
    }
    float bias0 = b2a[r], bias1 = b2a[16 + r];
#pragma unroll
    for (int v = 0; v < 8; ++v) {
        int M = v + 8 * half;
        Y[wave][M][r]      = fmaxf(c0[v] + bias0, 0.0f);
        Y[wave][M][16 + r] = fmaxf(c1[v] + bias1, 0.0f);
    }
    __syncthreads();   // covers smax init too

    // GEMM2: [16x32] @ [32x32]
    v8f d0 = {}; v8f d1 = {};
#pragma unroll
    for (int t = 0; t < 8; ++t) {
        int k0 = 4 * t + 2 * half;
        v2f a, b0, b1;
        a.x  = Y[wave][r][k0];
        a.y  = Y[wave][r][k0 + 1];
        b0.x = W2b[k0 * 32 + r];       b0.y = W2b[(k0 + 1) * 32 + r];
        b1.x = W2b[k0 * 32 + 16 + r];  b1.y = W2b[(k0 + 1) * 32 + 16 + r];
        d0 = __builtin_amdgcn_wmma_f32_16x16x4_f32(false, a, false, b0, (short)0, d0, false, false);
        d1 = __builtin_amdgcn_wmma_f32_16x16x4_f32(false, a, false, b1, (short)0, d1, false, false);
    }
    // relu(D + bias); each lane's 8 values share one column -> lane-local max.
    // relu outputs are >= 0 so 0 is a safe identity and the uint-bit atomicMax
    // trick preserves float ordering.
    float bias0b = b2b[r], bias1b = b2b[16 + r];
    float m0 = 0.0f, m1 = 0.0f;
#pragma unroll
    for (int v = 0; v < 8; ++v) {
        int M = v + 8 * half;
        if (base + M < nNodes) {
            m0 = fmaxf(m0, fmaxf(d0[v] + bias0b, 0.0f));
            m1 = fmaxf(m1, fmaxf(d1[v] + bias1b, 0.0f));
        }
    }
    atomicMax(&smax[r],      __float_as_uint(m0));
    atomicMax(&smax[16 + r], __float_as_uint(m1));
    __syncthreads();
    if (threadIdx.x < 32) atomicMax(&gmax[threadIdx.x], smax[threadIdx.x]);
}

// ------------------------- readout -----------------------------------------
__global__ void readout_kernel(const unsigned* __restrict__ gmax,
                               const float* __restrict__ Wm1, const float* __restrict__ bm1,
                               const float* __restrict__ Wm2, const float* __restrict__ bm2,
                               float* __restrict__ out) {
    if (threadIdx.x == 0 && blockIdx.x == 0) {
        float g[32];
        for (int j = 0; j < 32; ++j) g[j] = __uint_as_float(gmax[j]);
        float t[16];
        for (int i = 0; i < 16; ++i) {
            float s = bm1[i];
            for (int j = 0; j < 32; ++j) s += g[j] * Wm1[j * 16 + i];
            t[i] = fmaxf(s, 0.0f);
        }
        float l0 = bm2[0], l1 = bm2[1];
        for (int i = 0; i < 16; ++i) { l0 += t[i] * Wm2[i * 2]; l1 += t[i] * Wm2[i * 2 + 1]; }
        float mx = fmaxf(l0, l1);
        float e0 = __expf(l0 - mx), e1 = __expf(l1 - mx);
        float inv = 1.0f / (e0 + e1);
        out[0] = e0 * inv;
        out[1] = e1 * inv;
    }
}

// ---------------------------------------------------------------------------
extern "C" void kernel_launch(void* const* d_in, const int* in_sizes, int n_in,
                              void* d_out, int out_size, void* d_ws, size_t ws_size,
                              hipStream_t stream) {
    const float* node_feat = (const float*)d_in[0];
    const float* edge_feat = (const float*)d_in[1];
    const float* edge_mask = (const float*)d_in[2];
    const int*   src       = (const int*)d_in[3];
    const int*   dst       = (const int*)d_in[4];
    const float* W1a = (const float*)d_in[5];  const float* b1a = (const float*)d_in[6];
    const float* W1b = (const float*)d_in[7];  const float* b1b = (const float*)d_in[8];
    const float* W2a = (const float*)d_in[9];  const float* b2a = (const float*)d_in[10];
    const float* W2b = (const float*)d_in[11]; const float* b2b = (const float*)d_in[12];
    const float* Wm1 = (const float*)d_in[13]; const float* bm1 = (const float*)d_in[14];
    const float* Wm2 = (const float*)d_in[15]; const float* bm2 = (const float*)d_in[16];

    const int nNodes = in_sizes[0] / 7;
    const int nEdges = in_sizes[3];

    float* ws = (float*)d_ws;
    unsigned* gmax = (unsigned*)ws;                       // 32
    float* deg  = ws + 32;                                // N
    float* acc1 = ws + 32 + (long long)nNodes;            // N*12 (col 11 zero)
    float* acc2 = ws + 32 + (long long)nNodes * 13;       // N*32
    float* h1   = ws + 32 + (long long)nNodes * 45;       // N*32

    // zero gmax + deg + acc1 + acc2 (contiguous)
    long long nz = 32 + (long long)nNodes * 45;
    zero_kernel<<<2048, 256, 0, stream>>>(ws, nz);

    // layer 1
    edge1_kernel<<<2048, 256, 0, stream>>>(node_feat, edge_feat, edge_mask, src, dst,
                                           acc1, deg, nEdges);
    int mlpBlocks = (nNodes + 127) / 128;   // 8 waves/block, 16 nodes/wave
    mlp1_kernel<<<mlpBlocks, 256, 0, stream>>>(acc1, deg, W1a, b1a, W1b, b1b, h1, nNodes);

    // layer 2
    edge2_kernel<<<2048, 256, 0, stream>>>(h1, edge_mask, src, dst, acc2, nEdges);
    mlp2_kernel<<<mlpBlocks, 256, 0, stream>>>(acc2, deg, W2a, b2a, W2b, b2b, gmax, nNodes);

    // readout
    readout_kernel<<<1, 64, 0, stream>>>(gmax, Wm1, bm1, Wm2, bm2, (float*)d_out);
}